// WikiGraphSAGE_23124103922158
// MI455X (gfx1250) — compile-verified
//
#include <hip/hip_runtime.h>

typedef __attribute__((ext_vector_type(2))) float v2f;
typedef __attribute__((ext_vector_type(8))) float v8f;

#define IN_DIM 300
#define H_DIM  128
#define L_LAYERS 4

#define XS (IN_DIM + 2)   // padded LDS row stride for x tile (bank-conflict-free, 8B aligned)
#define HS (H_DIM + 4)    // padded LDS row stride for h/agg tiles

// ---------------- utility kernels ----------------

__global__ void zero_f32_kernel(float* __restrict__ p, int n) {
    int i = blockIdx.x * blockDim.x + threadIdx.x;
    if (i < n) p[i] = 0.0f;
}

__global__ void degree_kernel(const int* __restrict__ dst, float* __restrict__ deg, int e) {
    int i = blockIdx.x * blockDim.x + threadIdx.x;
    if (i < e) unsafeAtomicAdd(&deg[dst[i]], 1.0f);
}

// in[K][Nc] row-major -> out[Nc][K] row-major (so WMMA B loads are contiguous in K)
__global__ void transpose_kernel(const float* __restrict__ in, float* __restrict__ out,
                                 int K, int Nc) {
    int i = blockIdx.x * blockDim.x + threadIdx.x;
    if (i < K * Nc) {
        int k = i / Nc;
        int n = i - k * Nc;
        out[n * K + k] = in[i];
    }
}

// one wave per edge: lane moves 4 consecutive floats of the 128-wide row
__global__ __launch_bounds__(256)
void scatter_kernel(const int* __restrict__ src, const int* __restrict__ dst,
                    const float* __restrict__ h, float* __restrict__ agg, int e) {
    int wave = (blockIdx.x * blockDim.x + threadIdx.x) >> 5;
    int lane = threadIdx.x & 31;
    if (wave >= e) return;
    int s = src[wave];
    int d = dst[wave];
    const float4* hp = reinterpret_cast<const float4*>(h + (long)s * H_DIM);
    float4 v = hp[lane];                       // coalesced 512B per wave, L2-resident
    float* ap = agg + (long)d * H_DIM + lane * 4;
    unsafeAtomicAdd(ap + 0, v.x);
    unsafeAtomicAdd(ap + 1, v.y);
    unsafeAtomicAdd(ap + 2, v.z);
    unsafeAtomicAdd(ap + 3, v.w);
}

// ---------------- WMMA GEMM kernels (fp32, 16x16x4) ----------------

// h = relu(x @ W_emb + b_emb), one workgroup per 16-node tile, wave w -> cols [16w,16w+16)
__global__ __launch_bounds__(256)
void emb_kernel(const float* __restrict__ x, const float* __restrict__ WembT,
                const float* __restrict__ bemb, float* __restrict__ h, int n_nodes) {
    __shared__ float shx[16 * XS];
    const int  tid  = threadIdx.x;
    const long row0 = (long)blockIdx.x * 16;
    const bool full = (row0 + 16 <= (long)n_nodes);   // uniform across workgroup

    if (full) {
        for (int i = tid; i < 16 * IN_DIM; i += 256) {
            int r = i / IN_DIM, c = i - r * IN_DIM;
            shx[r * XS + c] = x[(row0 + r) * IN_DIM + c];
        }
    } else {
        for (int i = tid; i < 16 * IN_DIM; i += 256) {
            int r = i / IN_DIM, c = i - r * IN_DIM;
            long rr = row0 + r; if (rr >= n_nodes) rr = n_nodes - 1;
            shx[r * XS + c] = x[rr * IN_DIM + c];
        }
    }
    __syncthreads();

    const int wave = tid >> 5, lane = tid & 31;
    const int m = lane & 15, hi = lane >> 4;   // A row / B col = m; K-pair select = hi
    const int ncol = wave * 16 + m;

    v8f acc;
    float bv = bemb[ncol];
    #pragma unroll
    for (int i = 0; i < 8; ++i) acc[i] = bv;

    const float* wt = WembT + (long)ncol * IN_DIM;
    #pragma unroll 5
    for (int k0 = 0; k0 < IN_DIM; k0 += 4) {
        int k = k0 + 2 * hi;
        v2f a; a.x = shx[m * XS + k]; a.y = shx[m * XS + k + 1];
        v2f b; b.x = wt[k];           b.y = wt[k + 1];
        acc = __builtin_amdgcn_wmma_f32_16x16x4_f32(false, a, false, b,
                                                    (short)0, acc, false, false);
    }

    float* hrow = h + row0 * H_DIM + wave * 16 + m;
    if (full) {
        #pragma unroll
        for (int v = 0; v < 8; ++v) {
            float r = acc[v];
            hrow[(long)(v + 8 * hi) * H_DIM] = r > 0.0f ? r : 0.0f;
        }
    } else {
        #pragma unroll
        for (int v = 0; v < 8; ++v) {
            int mm = v + 8 * hi;
            if (row0 + mm < n_nodes) {
                float r = acc[v];
                hrow[(long)mm * H_DIM] = r > 0.0f ? r : 0.0f;
            }
        }
    }
}

// hout = relu((agg * inv_deg) @ W_l + b_l + hin @ W_r)
__global__ __launch_bounds__(256)
void sage_layer_kernel(const float* __restrict__ agg, const float* __restrict__ hin,
                       const float* __restrict__ deg,
                       const float* __restrict__ WlT, const float* __restrict__ WrT,
                       const float* __restrict__ bias, float* __restrict__ hout,
                       int n_nodes) {
    __shared__ float sh_a[16 * HS];
    __shared__ float sh_h[16 * HS];
    const int  tid  = threadIdx.x;
    const long row0 = (long)blockIdx.x * 16;
    const bool full = (row0 + 16 <= (long)n_nodes);   // uniform across workgroup

    // 256 threads load 16x128 of each matrix: thread t -> row t/128 of each pair, col t%128... use 2 rows/8 iters
    if (full) {
        for (int i = tid; i < 16 * H_DIM; i += 256) {
            int r = i >> 7, c = i & 127;
            long rr = row0 + r;
            float dg  = deg[rr];
            float inv = 1.0f / (dg > 1.0f ? dg : 1.0f);
            sh_a[r * HS + c] = agg[rr * H_DIM + c] * inv;  // fold mean-aggregation here
            sh_h[r * HS + c] = hin[rr * H_DIM + c];
        }
    } else {
        for (int i = tid; i < 16 * H_DIM; i += 256) {
            int r = i >> 7, c = i & 127;
            long rr = row0 + r; if (rr >= n_nodes) rr = n_nodes - 1;
            float dg  = deg[rr];
            float inv = 1.0f / (dg > 1.0f ? dg : 1.0f);
            sh_a[r * HS + c] = agg[rr * H_DIM + c] * inv;
            sh_h[r * HS + c] = hin[rr * H_DIM + c];
        }
    }
    __syncthreads();

    const int wave = tid >> 5, lane = tid & 31;
    const int m = lane & 15, hi = lane >> 4;
    const int ncol = wave * 16 + m;

    v8f acc;
    float bv = bias[ncol];
    #pragma unroll
    for (int i = 0; i < 8; ++i) acc[i] = bv;

    const float* wl = WlT + (long)ncol * H_DIM;
    const float* wr = WrT + (long)ncol * H_DIM;
    #pragma unroll 4
    for (int k0 = 0; k0 < H_DIM; k0 += 4) {
        int k = k0 + 2 * hi;
        v2f aa; aa.x = sh_a[m * HS + k]; aa.y = sh_a[m * HS + k + 1];
        v2f ah; ah.x = sh_h[m * HS + k]; ah.y = sh_h[m * HS + k + 1];
        v2f bl; bl.x = wl[k];            bl.y = wl[k + 1];
        v2f br; br.x = wr[k];            br.y = wr[k + 1];
        acc = __builtin_amdgcn_wmma_f32_16x16x4_f32(false, aa, false, bl,
                                                    (short)0, acc, false, false);
        acc = __builtin_amdgcn_wmma_f32_16x16x4_f32(false, ah, false, br,
                                                    (short)0, acc, false, false);
    }

    float* orow = hout + row0 * H_DIM + wave * 16 + m;
    if (full) {
        #pragma unroll
        for (int v = 0; v < 8; ++v) {
            float r = acc[v];
            orow[(long)(v + 8 * hi) * H_DIM] = r > 0.0f ? r : 0.0f;
        }
    } else {
        #pragma unroll
        for (int v = 0; v < 8; ++v) {
            int mm = v + 8 * hi;
            if (row0 + mm < n_nodes) {
                float r = acc[v];
                orow[(long)mm * H_DIM] = r > 0.0f ? r : 0.0f;
            }
        }
    }
}

// ---------------- host side ----------------

static inline int cdiv(long a, long b) { return (int)((a + b - 1) / b); }

extern "C" void kernel_launch(void* const* d_in, const int* in_sizes, int n_in,
                              void* d_out, int out_size, void* d_ws, size_t ws_size,
                              hipStream_t stream) {
    const float* x    = (const float*)d_in[0];
    const int*   ei   = (const int*)  d_in[1];   // [2,E] int32: row0=src, row1=dst
    const float* Wemb = (const float*)d_in[2];   // [300,128]
    const float* bemb = (const float*)d_in[3];   // [128]
    const float* Wl   = (const float*)d_in[4];   // [4,128,128]
    const float* bl   = (const float*)d_in[5];   // [4,128]
    const float* Wr   = (const float*)d_in[6];   // [4,128,128]
    float* out = (float*)d_out;

    const int N = in_sizes[0] / IN_DIM;
    const int E = in_sizes[1] / 2;
    const int* src = ei;
    const int* dst = ei + E;

    // workspace carve-out
    char*  ws  = (char*)d_ws;
    size_t off = 0;
    auto carve = [&](size_t bytes) -> float* {
        float* p = (float*)(ws + off);
        off = (off + bytes + 255) & ~(size_t)255;
        return p;
    };
    float* deg   = carve((size_t)N * sizeof(float));
    float* hA    = carve((size_t)N * H_DIM * sizeof(float));
    float* hB    = carve((size_t)N * H_DIM * sizeof(float));
    float* agg   = carve((size_t)N * H_DIM * sizeof(float));
    float* WembT = carve((size_t)IN_DIM * H_DIM * sizeof(float));
    float* WlT   = carve((size_t)L_LAYERS * H_DIM * H_DIM * sizeof(float));
    float* WrT   = carve((size_t)L_LAYERS * H_DIM * H_DIM * sizeof(float));
    (void)ws_size;

    const int tiles = cdiv(N, 16);

    // degree
    zero_f32_kernel<<<cdiv(N, 256), 256, 0, stream>>>(deg, N);
    degree_kernel<<<cdiv(E, 256), 256, 0, stream>>>(dst, deg, E);

    // weight transposes (tiny, one-time per call)
    transpose_kernel<<<cdiv((long)IN_DIM * H_DIM, 256), 256, 0, stream>>>(Wemb, WembT, IN_DIM, H_DIM);
    for (int l = 0; l < L_LAYERS; ++l) {
        transpose_kernel<<<cdiv((long)H_DIM * H_DIM, 256), 256, 0, stream>>>(
            Wl + (long)l * H_DIM * H_DIM, WlT + (long)l * H_DIM * H_DIM, H_DIM, H_DIM);
        transpose_kernel<<<cdiv((long)H_DIM * H_DIM, 256), 256, 0, stream>>>(
            Wr + (long)l * H_DIM * H_DIM, WrT + (long)l * H_DIM * H_DIM, H_DIM, H_DIM);
    }

    // input embedding
    emb_kernel<<<tiles, 256, 0, stream>>>(x, WembT, bemb, hA, N);

    // layers
    float* hcur = hA;
    for (int l = 0; l < L_LAYERS; ++l) {
        zero_f32_kernel<<<cdiv((long)N * H_DIM, 256), 256, 0, stream>>>(agg, N * H_DIM);
        scatter_kernel<<<cdiv(E, 8), 256, 0, stream>>>(src, dst, hcur, agg, E);
        float* hnext = (l == L_LAYERS - 1) ? out : (hcur == hA ? hB : hA);
        sage_layer_kernel<<<tiles, 256, 0, stream>>>(
            agg, hcur, deg,
            WlT + (long)l * H_DIM * H_DIM, WrT + (long)l * H_DIM * H_DIM,
            bl + (long)l * H_DIM, hnext, N);
        hcur = hnext;
    }
}